// Decoder_24137716203570
// MI455X (gfx1250) — compile-verified
//
#include <hip/hip_runtime.h>
#include <hip/hip_bf16.h>

// ---------------------------------------------------------------------------
// MI455X (gfx1250, wave32) implementation of the attention-LSTM decoder step.
// Workload is HBM-bound (~800MB moved, ~120 GFLOP), so all math stays in f32
// using V_WMMA_F32_16X16X4_F32 (full reference precision; f32 WMMA throughput
// vastly exceeds the ~3.5 TFLOP/s needed to keep up with 23.3 TB/s HBM).
// Data path: coalesced global_load_b128 -> LDS panels -> ds_load_b64 -> WMMA,
// plus global_prefetch_b8 on the 103MB logit_w stream.
// The K-slice inner loop has an unrolled kb==32 fast path: 16 ds_load_b64 in
// flight, one s_wait_dscnt, then 8 back-to-back v_wmma ops.
// ---------------------------------------------------------------------------

typedef __attribute__((ext_vector_type(2))) float v2f;
typedef __attribute__((ext_vector_type(8))) float v8f;

#define B_   1024
#define H_   512
#define S_   300
#define E_   512
#define V_   50257
#define NO_  36
#define NA_  26
#define NV_  26
#define NOS_ 20

// Inner product over one staged K-slice. aRow/bRow point at this lane's row
// base (already including the hi-half K offset). Unrolled fast path for the
// full 32-wide slice keeps the WMMA pipe fed from LDS.
__device__ __forceinline__ v8f kslice_mma(const float* __restrict__ aRow,
                                          const float* __restrict__ bRow,
                                          int kb, v8f c) {
    if (kb == 32) {
#pragma unroll
        for (int ks = 0; ks < 32; ks += 4) {
            v2f a = *(const v2f*)(aRow + ks);
            v2f b = *(const v2f*)(bRow + ks);
            c = __builtin_amdgcn_wmma_f32_16x16x4_f32(false, a, false, b,
                                                      (short)0, c, false, false);
        }
    } else {
        for (int ks = 0; ks < kb; ks += 4) {
            v2f a = *(const v2f*)(aRow + ks);
            v2f b = *(const v2f*)(bRow + ks);
            c = __builtin_amdgcn_wmma_f32_16x16x4_f32(false, a, false, b,
                                                      (short)0, c, false, false);
        }
    }
    return c;
}

// ---------------------------------------------------------------------------
// Generic NT GEMM: C[M,N] = A[M,K] * W[N,K]^T (+bias[n]) (+= existing C).
// Block: 256 threads = 8 waves. Tile: 16 rows x 128 cols (16 per wave).
// K sliced by 32 through LDS. M must be a multiple of 16; N tail guarded.
// WMMA f32 16x16x4 VGPR layout (ISA 7.12.2):
//   A: lanes 0-15 -> M=lane, regs {k,k+1}; lanes 16-31 -> M=lane-16, {k+2,k+3}
//   B: mirrored (lanes = N, reg pair = same K split)
//   C: vgpr r -> M = r + 8*(lane>=16), N = lane%16
// ---------------------------------------------------------------------------
__global__ __launch_bounds__(256)
void gemm_nt_kernel(const float* __restrict__ A, const float* __restrict__ W,
                    float* __restrict__ C, const float* __restrict__ bias,
                    int M, int N, int K, int lda, int ldw, int ldc, int accum) {
    const int m0  = blockIdx.y * 16;
    const int n0  = blockIdx.x * 128;
    const int tid = threadIdx.x;
    const int lane = tid & 31, wave = tid >> 5;
    const int hi = lane >> 4, lm = lane & 15;
    const int koff = hi ? 2 : 0;

    __shared__ float As[16 * 34];   // 16 rows x 32 k (+2 pad, keeps 8B align)
    __shared__ float Bs[128 * 34];

    const float* aRow = &As[lm * 34 + koff];
    const float* bRow = &Bs[(wave * 16 + lm) * 34 + koff];

    v8f c = {};
    for (int k0 = 0; k0 < K; k0 += 32) {
        const int kb = min(32, K - k0);
        __syncthreads();
        // stage A panel (16 x kb), coalesced float4
        for (int idx = tid; idx < 16 * 8; idx += 256) {
            int r = idx >> 3, q = idx & 7;
            if (q * 4 < kb) {
                float4 v = *(const float4*)(A + (size_t)(m0 + r) * lda + k0 + q * 4);
                float* d = &As[r * 34 + q * 4];
                d[0] = v.x; d[1] = v.y; d[2] = v.z; d[3] = v.w;
            }
        }
        // stage B panel (128 x kb), coalesced float4, N tail skipped (stores guarded)
        for (int idx = tid; idx < 128 * 8; idx += 256) {
            int r = idx >> 3, q = idx & 7;
            int n = n0 + r;
            if (q * 4 < kb && n < N) {
                const float* src = W + (size_t)n * ldw + k0 + q * 4;
                float4 v = *(const float4*)src;
                float* d = &Bs[r * 34 + q * 4];
                d[0] = v.x; d[1] = v.y; d[2] = v.z; d[3] = v.w;
                if (q == 0 && k0 + 32 < K)                // stream next K-slice
                    __builtin_prefetch(src + 32, 0, 1);   // -> global_prefetch_b8
            }
        }
        __syncthreads();
        c = kslice_mma(aRow, bRow, kb, c);
    }
    const int n = n0 + wave * 16 + lm;
    if (n < N) {
        const float bv = bias ? bias[n] : 0.0f;
#pragma unroll
        for (int r = 0; r < 8; ++r) {
            const size_t row = (size_t)(m0 + r + 8 * hi);
            float v = c[r] + bv;
            if (accum) v += C[row * ldc + n];
            C[row * ldc + n] = v;
        }
    }
}

// ---------------------------------------------------------------------------
// Fused attention-score kernel: for flattened rows m = b*Nper + n,
//   scores[m] = sum_h wvec[h] * tanh( (feats @ Uw^T)[m,h] + Wh[b,h] + biasU[h] )
// One block per 16 rows; full A panel (16 x K <= 32KB) staged once in LDS;
// H=512 covered in four 128-col chunks; epilogue does tanh + shfl_xor butterfly
// reduction over N within each 16-lane half, accumulated via LDS atomics.
// ---------------------------------------------------------------------------
__global__ __launch_bounds__(256)
void attn_score_kernel(const float* __restrict__ feats, const float* __restrict__ Uw,
                       const float* __restrict__ biasU, const float* __restrict__ wvec,
                       const float* __restrict__ Wh, float* __restrict__ scores,
                       int Nper, int K) {
    const int m0  = blockIdx.x * 16;
    const int tid = threadIdx.x;
    const int lane = tid & 31, wave = tid >> 5;
    const int hi = lane >> 4, lm = lane & 15;
    const int koff = hi ? 2 : 0;
    const int AS = K + 2;                 // even row stride -> 8B-aligned v2f reads

    __shared__ float Af[16 * 514];        // full 16 x K panel (K <= 512)
    __shared__ float Bs[128 * 34];
    __shared__ float sacc[16];

    if (tid < 16) sacc[tid] = 0.0f;
    const int kq = K >> 2;
    for (int idx = tid; idx < 16 * kq; idx += 256) {
        int r = idx / kq, q = idx - r * kq;
        float4 v = *(const float4*)(feats + (size_t)(m0 + r) * K + q * 4);
        float* d = &Af[r * AS + q * 4];
        d[0] = v.x; d[1] = v.y; d[2] = v.z; d[3] = v.w;
    }
    __syncthreads();

    const float* bRow = &Bs[(wave * 16 + lm) * 34 + koff];

    for (int n0 = 0; n0 < H_; n0 += 128) {
        v8f c = {};
        for (int k0 = 0; k0 < K; k0 += 32) {
            const int kb = min(32, K - k0);
            __syncthreads();
            for (int idx = tid; idx < 128 * 8; idx += 256) {
                int r = idx >> 3, q = idx & 7;
                if (q * 4 < kb) {
                    float4 v = *(const float4*)(Uw + (size_t)(n0 + r) * K + k0 + q * 4);
                    float* d = &Bs[r * 34 + q * 4];
                    d[0] = v.x; d[1] = v.y; d[2] = v.z; d[3] = v.w;
                }
            }
            __syncthreads();
            c = kslice_mma(&Af[lm * AS + k0 + koff], bRow, kb, c);
        }
        // epilogue: per-element tanh, weight, reduce over the 16 columns held
        // by this half-wave, accumulate per-row partials across chunks/waves.
        const int h = n0 + wave * 16 + lm;
        const float wv = wvec[h];
        const float bu = biasU[h];
#pragma unroll
        for (int r = 0; r < 8; ++r) {
            const int m   = r + 8 * hi;
            const int row = m0 + m;
            const int bi  = row / Nper;
            float p = wv * tanhf(c[r] + Wh[(size_t)bi * H_ + h] + bu);
            p += __shfl_xor(p, 1);
            p += __shfl_xor(p, 2);
            p += __shfl_xor(p, 4);
            p += __shfl_xor(p, 8);
            if (lm == 0) atomicAdd(&sacc[m], p);   // ds_add_f32
        }
    }
    __syncthreads();
    if (tid < 16) scores[m0 + tid] = sacc[tid];
}

// softmax over Nper scores + weighted sum of feats -> dest[:, coloff:coloff+D]
__global__ __launch_bounds__(256)
void attn_apply_kernel(const float* __restrict__ scores, const float* __restrict__ feats,
                       float* __restrict__ dest, int Nper, int D, int ldd, int coloff) {
    const int b = blockIdx.x, tid = threadIdx.x;
    __shared__ float sa[64];
    __shared__ float sw[64];
    if (tid < Nper) sa[tid] = scores[(size_t)b * Nper + tid];
    __syncthreads();
    float mx = -1e30f;
    for (int n = 0; n < Nper; ++n) mx = fmaxf(mx, sa[n]);
    float s = 0.0f;
    for (int n = 0; n < Nper; ++n) s += __expf(sa[n] - mx);
    if (tid < Nper) sw[tid] = __expf(sa[tid] - mx) / s;
    __syncthreads();
    for (int d = tid; d < D; d += 256) {
        float acc = 0.0f;
        for (int n = 0; n < Nper; ++n)
            acc += sw[n] * feats[((size_t)b * Nper + n) * D + d];
        dest[(size_t)b * ldd + coloff + d] = acc;
    }
}

// fill semcat[:,300:900] from action/video semantics and x[:,1024:1536] from embed
__global__ __launch_bounds__(512)
void prep_kernel(const float* __restrict__ act_sem, const float* __restrict__ vid_sem,
                 const float* __restrict__ embed, float* __restrict__ semcat,
                 float* __restrict__ x) {
    const int b = blockIdx.x, t = threadIdx.x;
    if (t < S_) {
        semcat[(size_t)b * 900 + 300 + t] = act_sem[(size_t)b * S_ + t];
        semcat[(size_t)b * 900 + 600 + t] = vid_sem[(size_t)b * S_ + t];
    }
    x[(size_t)b * 1536 + 1024 + t] = embed[(size_t)b * E_ + t];
}

__device__ __forceinline__ float sigf(float x) { return 1.0f / (1.0f + __expf(-x)); }

// LSTM cell (torch gate order i,f,g,o), writes h/c directly into d_out tail
__global__ __launch_bounds__(256)
void lstm_kernel(const float* __restrict__ gates, const float* __restrict__ c0,
                 float* __restrict__ h_out, float* __restrict__ c_out) {
    const int idx = blockIdx.x * 256 + threadIdx.x;     // B*H total
    const int b = idx >> 9, hh = idx & 511;
    const float* g = gates + (size_t)b * 2048;
    const float i = g[hh], f = g[512 + hh], gg = g[1024 + hh], o = g[1536 + hh];
    const float c = sigf(f) * c0[idx] + sigf(i) * tanhf(gg);
    h_out[idx] = sigf(o) * tanhf(c);
    c_out[idx] = c;
}

// in-place row-wise log_softmax via online logsumexp (one block per row)
__global__ __launch_bounds__(256)
void logsoftmax_kernel(float* __restrict__ logits, int V) {
    const int b = blockIdx.x, tid = threadIdx.x;
    float* row = logits + (size_t)b * V;
    float m = -3.0e38f, s = 0.0f;
    for (int j = tid; j < V; j += 256) {
        const float x = row[j];
        if (x > m) { s = s * __expf(m - x) + 1.0f; m = x; }
        else       { s += __expf(x - m); }
    }
    __shared__ float sm[256], ss[256];
    sm[tid] = m; ss[tid] = s;
    __syncthreads();
    for (int off = 128; off > 0; off >>= 1) {
        if (tid < off) {
            const float m1 = sm[tid], s1 = ss[tid];
            const float m2 = sm[tid + off], s2 = ss[tid + off];
            const float mm = fmaxf(m1, m2);
            ss[tid] = s1 * __expf(m1 - mm) + s2 * __expf(m2 - mm);
            sm[tid] = mm;
        }
        __syncthreads();
    }
    const float lse = sm[0] + logf(ss[0]);
    for (int j = tid; j < V; j += 256) row[j] -= lse;
}

extern "C" void kernel_launch(void* const* d_in, const int* in_sizes, int n_in,
                              void* d_out, int out_size, void* d_ws, size_t ws_size,
                              hipStream_t stream) {
    const float* objects  = (const float*)d_in[0];
    const float* action   = (const float*)d_in[1];
    const float* video    = (const float*)d_in[2];
    const float* obj_sem  = (const float*)d_in[3];
    const float* act_sem  = (const float*)d_in[4];
    const float* vid_sem  = (const float*)d_in[5];
    const float* embed    = (const float*)d_in[6];
    const float* h0       = (const float*)d_in[7];
    const float* c0       = (const float*)d_in[8];
    const float* W_w      = (const float*)d_in[9];
    const float* Uo_w     = (const float*)d_in[10];
    const float* bo       = (const float*)d_in[11];
    const float* wo_w     = (const float*)d_in[12];
    const float* Uos_w    = (const float*)d_in[13];
    const float* bos      = (const float*)d_in[14];
    const float* wos_w    = (const float*)d_in[15];
    const float* Um_w     = (const float*)d_in[16];
    const float* bm       = (const float*)d_in[17];
    const float* wm_w     = (const float*)d_in[18];
    const float* Uv_w     = (const float*)d_in[19];
    const float* bv       = (const float*)d_in[20];
    const float* wv_w     = (const float*)d_in[21];
    const float* vis_w    = (const float*)d_in[22];
    const float* vis_b    = (const float*)d_in[23];
    const float* sem_w    = (const float*)d_in[24];
    const float* sem_b    = (const float*)d_in[25];
    const float* W_ih     = (const float*)d_in[26];
    const float* W_hh     = (const float*)d_in[27];
    const float* b_ih     = (const float*)d_in[28];
    const float* b_hh     = (const float*)d_in[29];
    const float* to_word_w = (const float*)d_in[30];
    const float* to_word_b = (const float*)d_in[31];
    const float* logit_w  = (const float*)d_in[32];
    const float* logit_b  = (const float*)d_in[33];

    float* out = (float*)d_out;              // [B*V logits | B*H h | B*H c]
    float* h_out = out + (size_t)B_ * V_;
    float* c_out = h_out + (size_t)B_ * H_;

    // workspace layout (floats), ~29 MB total
    float* ws     = (float*)d_ws;
    float* wh     = ws;                              // [B,512]
    float* viscat = wh     + (size_t)B_ * H_;        // [B,1536] video|motion|objects
    float* semcat = viscat + (size_t)B_ * 1536;      // [B,900]  objs|act|vid
    float* x      = semcat + (size_t)B_ * 900;       // [B,1536] visual|sem|embed
    float* gates  = x      + (size_t)B_ * 1536;      // [B,2048]
    float* word   = gates  + (size_t)B_ * 2048;      // [B,512]
    float* sO     = word   + (size_t)B_ * H_;        // [B*36]
    float* sM     = sO + (size_t)B_ * NO_;           // [B*26]
    float* sV     = sM + (size_t)B_ * NA_;           // [B*26]
    float* sOS    = sV + (size_t)B_ * NV_;           // [B*20]

    // 0) concat fills
    prep_kernel<<<B_, 512, 0, stream>>>(act_sem, vid_sem, embed, semcat, x);

    // 1) Wh = h0 @ W_w^T
    gemm_nt_kernel<<<dim3(4, B_ / 16), 256, 0, stream>>>(
        h0, W_w, wh, nullptr, B_, H_, H_, H_, H_, H_, 0);

    // 2) fused attention scores (GEMM + tanh + dot-reduce)
    attn_score_kernel<<<(B_ * NO_) / 16, 256, 0, stream>>>(
        objects, Uo_w, bo, wo_w, wh, sO, NO_, H_);
    attn_score_kernel<<<(B_ * NA_) / 16, 256, 0, stream>>>(
        action, Um_w, bm, wm_w, wh, sM, NA_, H_);
    attn_score_kernel<<<(B_ * NV_) / 16, 256, 0, stream>>>(
        video, Uv_w, bv, wv_w, wh, sV, NV_, H_);
    attn_score_kernel<<<(B_ * NOS_) / 16, 256, 0, stream>>>(
        obj_sem, Uos_w, bos, wos_w, wh, sOS, NOS_, S_);

    // 3) softmax + weighted sums into concat buffers
    attn_apply_kernel<<<B_, 256, 0, stream>>>(sV, video,   viscat, NV_, H_, 1536, 0);
    attn_apply_kernel<<<B_, 256, 0, stream>>>(sM, action,  viscat, NA_, H_, 1536, 512);
    attn_apply_kernel<<<B_, 256, 0, stream>>>(sO, objects, viscat, NO_, H_, 1536, 1024);
    attn_apply_kernel<<<B_, 256, 0, stream>>>(sOS, obj_sem, semcat, NOS_, S_, 900, 0);

    // 4) visual / sem projections into x[:,0:512] and x[:,512:1024]
    gemm_nt_kernel<<<dim3(4, B_ / 16), 256, 0, stream>>>(
        viscat, vis_w, x, vis_b, B_, H_, 1536, 1536, 1536, 1536, 0);
    gemm_nt_kernel<<<dim3(4, B_ / 16), 256, 0, stream>>>(
        semcat, sem_w, x + 512, sem_b, B_, H_, 900, 900, 900, 1536, 0);

    // 5) gates = x @ W_ih^T + b_ih  +  h0 @ W_hh^T + b_hh
    gemm_nt_kernel<<<dim3(16, B_ / 16), 256, 0, stream>>>(
        x, W_ih, gates, b_ih, B_, 2048, 1536, 1536, 1536, 2048, 0);
    gemm_nt_kernel<<<dim3(16, B_ / 16), 256, 0, stream>>>(
        h0, W_hh, gates, b_hh, B_, 2048, H_, H_, H_, 2048, 1);

    // 6) LSTM cell -> h/c regions of d_out
    lstm_kernel<<<(B_ * H_) / 256, 256, 0, stream>>>(gates, c0, h_out, c_out);

    // 7) word = h @ to_word_w^T + b
    gemm_nt_kernel<<<dim3(4, B_ / 16), 256, 0, stream>>>(
        h_out, to_word_w, word, to_word_b, B_, E_, H_, H_, H_, E_, 0);

    // 8) logits = word @ logit_w^T + b  (N=50257 tail guarded), then log_softmax
    gemm_nt_kernel<<<dim3((V_ + 127) / 128, B_ / 16), 256, 0, stream>>>(
        word, logit_w, out, logit_b, B_, V_, E_, E_, E_, V_, 0);
    logsoftmax_kernel<<<B_, 256, 0, stream>>>(out, V_);
}